// SoftDTWBatch_90031104459524
// MI455X (gfx1250) — compile-verified
//
#include <hip/hip_runtime.h>
#include <hip/hip_bf16.h>
#include <math.h>

#define BATCH 64
#define TLEN  512
#define DIM   16
#define BIGF  1e10f

typedef __attribute__((ext_vector_type(2))) float v2f;
typedef __attribute__((ext_vector_type(8))) float v8f;

// ---------------------------------------------------------------------------
// Kernel 0: per-row squared norms ||x_t||^2, ||y_s||^2  (B*T rows of 16 floats)
// ---------------------------------------------------------------------------
__global__ void sdtw_norms(const float* __restrict__ x, const float* __restrict__ y,
                           float* __restrict__ x2, float* __restrict__ y2) {
  int idx = blockIdx.x * blockDim.x + threadIdx.x;
  if (idx >= BATCH * TLEN) return;
  const float4* xr = (const float4*)(x + (size_t)idx * DIM);
  const float4* yr = (const float4*)(y + (size_t)idx * DIM);
  float sx = 0.f, sy = 0.f;
#pragma unroll
  for (int q = 0; q < 4; ++q) {
    float4 a = xr[q]; float4 b = yr[q];
    sx += a.x*a.x + a.y*a.y + a.z*a.z + a.w*a.w;
    sy += b.x*b.x + b.y*b.y + b.z*b.z + b.w*b.w;
  }
  x2[idx] = sx;
  y2[idx] = sy;
}

// ---------------------------------------------------------------------------
// Kernel 1: D[b,t,s] = max(0, x2[t] + y2[s] - 2*<x_t, y_s>) via fp32 WMMA.
// One wave per 16x16 output tile; K=16 -> four V_WMMA_F32_16X16X4_F32 steps.
//
// ISA layout (16x4 fp32 A): lanes 0-15 hold M=0..15, lanes 16-31 also M=0..15;
// vgpr0 = {K=0 (lo lanes), K=2 (hi lanes)}, vgpr1 = {K=1, K=3}.
// B (4x16) mirrors this with N in place of M; since B = y^T and y is stored
// row-major (s, d), the per-lane load pattern is identical for A and B.
// ---------------------------------------------------------------------------
__global__ void sdtw_dist(const float* __restrict__ x, const float* __restrict__ y,
                          const float* __restrict__ x2, const float* __restrict__ y2,
                          float* __restrict__ Dm) {
  const int lane = threadIdx.x;                       // 0..31
  const int wave = threadIdx.y;                       // 0..3
  const int b    = blockIdx.z;
  const int row0 = blockIdx.y * 16;                   // tile row (t)
  const int col0 = (blockIdx.x * 4 + wave) * 16;      // tile col (s)
  const int hi   = lane >> 4;
  const int lm   = lane & 15;

  const float* xr = x + ((size_t)b * TLEN + row0 + lm) * DIM;  // A row M=lm
  const float* yr = y + ((size_t)b * TLEN + col0 + lm) * DIM;  // B col N=lm

  v8f acc = {0.f, 0.f, 0.f, 0.f, 0.f, 0.f, 0.f, 0.f};
#pragma unroll
  for (int ks = 0; ks < DIM; ks += 4) {
    v2f av, bv;
    av.x = xr[ks + 2 * hi + 0];
    av.y = xr[ks + 2 * hi + 1];
    bv.x = yr[ks + 2 * hi + 0];
    bv.y = yr[ks + 2 * hi + 1];
    acc = __builtin_amdgcn_wmma_f32_16x16x4_f32(
        /*neg_a=*/false, av, /*neg_b=*/false, bv,
        /*c_mod=*/(short)0, acc, /*reuse_a=*/false, /*reuse_b=*/false);
  }

  // C/D layout: vgpr r -> row M = r + 8*hi, col N = lm.
  const float y2v = y2[b * TLEN + col0 + lm];
  float* drow = Dm + ((size_t)b * TLEN + row0 + 8 * hi) * TLEN + col0 + lm;
#pragma unroll
  for (int r = 0; r < 8; ++r) {
    float xv = x2[b * TLEN + row0 + 8 * hi + r];
    float d  = xv + y2v - 2.0f * acc[r];
    drow[(size_t)r * TLEN] = d > 0.f ? d : 0.f;
  }
}

// ---------------------------------------------------------------------------
// Kernel 2: soft-DTW wavefront DP. One workgroup per batch, thread t owns DP
// row i = t+1 for the whole sweep, so each thread streams its own D row
// sequentially (cacheline reuse across anti-diagonal steps). Three rotating
// diagonals live in LDS (3 x 513 floats).
//   R[i][j] = D[i-1][j-1] + softmin(R[i-1][j-1], R[i-1][j], R[i][j-1])
// with R[0][0]=0, borders BIG; answer = R[T][T].  gamma = 1.
// ---------------------------------------------------------------------------
__global__ void sdtw_dp(const float* __restrict__ Dm, float* __restrict__ out) {
  __shared__ float buf[3][TLEN + 1];
  const int b = blockIdx.x;
  const int t = threadIdx.x;      // 0..511
  const int i = t + 1;            // DP row 1..512
  const float* Db = Dm + (size_t)b * TLEN * TLEN + (size_t)(i - 1) * TLEN;

  // diag k=0: only (0,0)=0; diag k=1: (0,1),(1,0)=BIG. Scrub everything else.
  for (int p = t; p <= TLEN; p += TLEN) {
    buf[0][p] = (p == 0) ? 0.f : BIGF;
    buf[1][p] = BIGF;
    buf[2][p] = BIGF;
  }
  __syncthreads();

  for (int k = 2; k <= 2 * TLEN; ++k) {
    float*       cur = buf[k % 3];
    const float* p1  = buf[(k + 2) % 3];   // diag k-1
    const float* p2  = buf[(k + 1) % 3];   // diag k-2
    int ilo = k - TLEN; if (ilo < 1) ilo = 1;
    int ihi = k - 1;    if (ihi > TLEN) ihi = TLEN;
    if (i >= ilo && i <= ihi) {
      float d = Db[k - i - 1];             // D[i-1][j-1], j = k-i
      float a = p2[i - 1];                 // R[i-1][j-1]
      float u = p1[i - 1];                 // R[i-1][j]
      float l = p1[i];                     // R[i][j-1]
      float m = fminf(a, fminf(u, l));
      float s = __expf(m - a) + __expf(m - u) + __expf(m - l);
      cur[i] = d + m - __logf(s);          // gamma == 1
    }
    if (t == 0 && k <= TLEN) {             // borders R[0][k], R[k][0]
      cur[0] = BIGF;
      cur[k] = BIGF;
    }
    __syncthreads();
  }
  if (t == 0) out[b] = buf[(2 * TLEN) % 3][TLEN];   // R[T][T]
}

// ---------------------------------------------------------------------------
extern "C" void kernel_launch(void* const* d_in, const int* in_sizes, int n_in,
                              void* d_out, int out_size, void* d_ws, size_t ws_size,
                              hipStream_t stream) {
  const float* x = (const float*)d_in[0];
  const float* y = (const float*)d_in[1];
  float* out = (float*)d_out;

  char*  ws = (char*)d_ws;
  float* Dm = (float*)ws;                                               // 64 MB
  float* x2 = (float*)(ws + (size_t)BATCH * TLEN * TLEN * sizeof(float));
  float* y2 = x2 + BATCH * TLEN;

  sdtw_norms<<<dim3((BATCH * TLEN + 255) / 256), dim3(256), 0, stream>>>(x, y, x2, y2);
  sdtw_dist<<<dim3(TLEN / 16 / 4, TLEN / 16, BATCH), dim3(32, 4), 0, stream>>>(x, y, x2, y2, Dm);
  sdtw_dp<<<dim3(BATCH), dim3(TLEN), 0, stream>>>(Dm, out);
}